// GraphNorm_32796370273054
// MI455X (gfx1250) — compile-verified
//
#include <hip/hip_runtime.h>

#define N_NODES   1000000
#define N_FEAT    128
#define N_GRAPHS  1024
#define EPS_F     1e-5f

typedef __attribute__((ext_vector_type(2))) float v2f;
typedef __attribute__((ext_vector_type(8))) float v8f;

// ws layout (floats):
//  [0,1024)     gn   : per-graph element counts (counts*F)
//  [1024,2048)  gs   : per-graph sum(x)
//  [2048,3072)  gss  : per-graph sum(x^2)
//  [3072,4096)  mean
//  [4096,5120)  istd

__global__ void gn_zero_kernel(float* ws) {
    int i = blockIdx.x * blockDim.x + threadIdx.x;
    if (i < 3 * N_GRAPHS) ws[i] = 0.0f;
}

// Pass 1: per-wave 16-node tile. WMMA f32 16x16x4:
//   csum = A x ones  -> every column = per-node feature sum
//   css  = A x A     -> (A and B register layouts coincide) Gram matrix,
//                       diagonal = per-node sum of squares
__global__ __launch_bounds__(256) void gn_reduce_kernel(
    const float* __restrict__ x, const int* __restrict__ batch,
    float* __restrict__ gn, float* __restrict__ gs, float* __restrict__ gss) {
    __shared__ float s_n[N_GRAPHS];
    __shared__ float s_s[N_GRAPHS];
    __shared__ float s_q[N_GRAPHS];
    __shared__ int sh_gf, sh_span;

    const int tid = threadIdx.x;
    const int nodeFirst = blockIdx.x * 128;   // 8 waves * 16 nodes
    if (tid == 0) {
        int nodeLast = nodeFirst + 127;
        if (nodeLast >= N_NODES) nodeLast = N_NODES - 1;
        int gf = batch[nodeFirst];
        int gl = batch[nodeLast];
        sh_gf = gf;
        sh_span = gl - gf + 1;
    }
    __syncthreads();
    const int gf = sh_gf;
    const int span = sh_span;
    for (int i = tid; i < span; i += 256) { s_n[i] = 0.f; s_s[i] = 0.f; s_q[i] = 0.f; }
    __syncthreads();

    const int wave = tid >> 5;
    const int lane = tid & 31;
    const int tileBase = nodeFirst + wave * 16;
    if (tileBase < N_NODES) {                 // wave-uniform guard (EXEC all-1 for WMMA)
        const int half = lane >> 4;           // K-half: lanes 0-15 -> K{0,1}, 16-31 -> K{2,3}
        const int mrow = lane & 15;           // node row within tile
        const float* rowp = x + (size_t)(tileBase + mrow) * N_FEAT + half * 2;
        v2f ones; ones.x = 1.0f; ones.y = 1.0f;
        v8f csum = {};
        v8f css  = {};
        #pragma unroll
        for (int kc = 0; kc < 32; ++kc) {     // 32 chunks of 4 features = 128
            v2f a = *(const v2f*)(rowp + kc * 4);
            csum = __builtin_amdgcn_wmma_f32_16x16x4_f32(false, a, false, ones,
                                                         (short)0, csum, false, false);
            css  = __builtin_amdgcn_wmma_f32_16x16x4_f32(false, a, false, a,
                                                         (short)0, css,  false, false);
        }
        // C/D layout: VGPR j holds M=j (lanes 0-15, N=lane) and M=j+8 (lanes 16-31, N=lane-16).
        // node j   (j<8): sum & diag at (vgpr j, lane j)
        // node 8+j      : sum & diag at (vgpr j, lane 24+j)
        const bool active = (lane < 8) || (lane >= 24);
        if (active) {
            const int j = (lane < 8) ? lane : (lane - 24);
            const int node = tileBase + ((lane < 8) ? lane : (lane - 16));
            float s = 0.f, q = 0.f;
            #pragma unroll
            for (int e = 0; e < 8; ++e) {
                if (j == e) { s = csum[e]; q = css[e]; }
            }
            const int slot = batch[node] - gf;
            atomicAdd(&s_s[slot], s);
            atomicAdd(&s_q[slot], q);
            atomicAdd(&s_n[slot], (float)N_FEAT);
        }
    }
    __syncthreads();
    for (int i = tid; i < span; i += 256) {
        float n = s_n[i];
        if (n != 0.f) {
            atomicAdd(&gn[gf + i],  n);
            atomicAdd(&gs[gf + i],  s_s[i]);
            atomicAdd(&gss[gf + i], s_q[i]);
        }
    }
}

__global__ void gn_finalize_kernel(const float* __restrict__ gn,
                                   const float* __restrict__ gs,
                                   const float* __restrict__ gss,
                                   float* __restrict__ gmean,
                                   float* __restrict__ gistd) {
    int g = blockIdx.x * blockDim.x + threadIdx.x;
    if (g < N_GRAPHS) {
        float denom = fmaxf(gn[g], 1.0f);
        float mean  = gs[g] / denom;
        float var   = fmaxf(gss[g] / denom - mean * mean, 0.0f);
        gmean[g] = mean;
        gistd[g] = rsqrtf(var + EPS_F);
    }
}

// Pass 3: coalesced float4 streaming normalize.
__global__ __launch_bounds__(256) void gn_norm_kernel(
    const float4* __restrict__ x4, const int* __restrict__ batch,
    const float4* __restrict__ w4, const float4* __restrict__ b4,
    const float* __restrict__ gmean, const float* __restrict__ gistd,
    float4* __restrict__ out4) {
    const int idx = blockIdx.x * 256 + threadIdx.x;   // < 32M, fits int
    const int node = idx >> 5;                        // 32 float4 per row
    const int col  = idx & 31;
    const int g = batch[node];
    const float m = gmean[g];
    const float is = gistd[g];
    float4 xv = x4[idx];
    float4 wv = w4[col];
    float4 bv = b4[col];
    float4 o;
    o.x = wv.x * (xv.x - m) * is + bv.x;
    o.y = wv.y * (xv.y - m) * is + bv.y;
    o.z = wv.z * (xv.z - m) * is + bv.z;
    o.w = wv.w * (xv.w - m) * is + bv.w;
    out4[idx] = o;
}

extern "C" void kernel_launch(void* const* d_in, const int* in_sizes, int n_in,
                              void* d_out, int out_size, void* d_ws, size_t ws_size,
                              hipStream_t stream) {
    const float* x      = (const float*)d_in[0];
    const int*   batch  = (const int*)d_in[1];
    const float* weight = (const float*)d_in[2];
    const float* bias   = (const float*)d_in[3];
    float* out = (float*)d_out;

    float* ws    = (float*)d_ws;
    float* gn    = ws;
    float* gs    = ws + N_GRAPHS;
    float* gss   = ws + 2 * N_GRAPHS;
    float* gmean = ws + 3 * N_GRAPHS;
    float* gistd = ws + 4 * N_GRAPHS;

    // zero accumulators every call (determinism: atomics accumulate)
    gn_zero_kernel<<<(3 * N_GRAPHS + 255) / 256, 256, 0, stream>>>(ws);

    const int reduceBlocks = (N_NODES + 127) / 128;   // 7813
    gn_reduce_kernel<<<reduceBlocks, 256, 0, stream>>>(x, batch, gn, gs, gss);

    gn_finalize_kernel<<<(N_GRAPHS + 255) / 256, 256, 0, stream>>>(gn, gs, gss, gmean, gistd);

    const int nF4 = N_NODES * (N_FEAT / 4);           // 32,000,000
    gn_norm_kernel<<<nF4 / 256, 256, 0, stream>>>(
        (const float4*)x, batch, (const float4*)weight, (const float4*)bias,
        gmean, gistd, (float4*)out);
}